// MultiDomainMultiCriteriaClassifier_16363825397829
// MI455X (gfx1250) — compile-verified
//
#include <hip/hip_runtime.h>
#include <hip/hip_bf16.h>
#include <math.h>

typedef __attribute__((ext_vector_type(2))) float v2f;
typedef __attribute__((ext_vector_type(8))) float v8f;

#define BB 16384
#define PP 32
#define LL 16
#define DD 768
#define HH 8

// ---------------------------------------------------------------------------
// Kernel 1: S2[p,l] = dot(crit_emb[p,l,:], W[head_map[p,l],:]) + bias[head]
// One wave per (p,l). 512 waves total; trivial cost (~0.4 MFLOP).
// ---------------------------------------------------------------------------
__global__ void mdmc_s2_kernel(const float* __restrict__ crit_emb,
                               const float* __restrict__ W,
                               const float* __restrict__ bias,
                               const int* __restrict__ head_map,
                               float* __restrict__ s2) {
    int gwave = (blockIdx.x * blockDim.x + threadIdx.x) >> 5;
    int lane  = threadIdx.x & 31;
    if (gwave >= PP * LL) return;

    int h = head_map[gwave];
    const float* ce = crit_emb + (size_t)gwave * DD;
    const float* wv = W + (size_t)h * DD;

    float acc = 0.0f;
    #pragma unroll 4
    for (int d = lane; d < DD; d += 32)
        acc = fmaf(ce[d], wv[d], acc);

    // wave32 butterfly reduction
    #pragma unroll
    for (int off = 16; off > 0; off >>= 1)
        acc += __shfl_xor(acc, off, 32);

    if (lane == 0)
        s2[gwave] = acc + bias[h];
}

// ---------------------------------------------------------------------------
// Kernel 2: S1 = emb @ W^T via v_wmma_f32_16x16x4_f32, fused epilogue.
// Block = 128 threads = 4 waves; each wave owns a 16-sample tile.
// W^T (zero-padded to 16 head columns) lives in LDS; embeddings are streamed
// once from HBM (the bandwidth-bound part: 50.3 MB total).
// ---------------------------------------------------------------------------
__global__ void __launch_bounds__(128)
mdmc_main_kernel(const float* __restrict__ emb,
                 const float* __restrict__ W,
                 const int* __restrict__ pidx,
                 const int* __restrict__ head_map,
                 const int* __restrict__ crit_len,
                 const float* __restrict__ s2,
                 float* __restrict__ out) {
    __shared__ float ldsW[DD * 16];        // 48 KB: W^T padded, [d][16 cols]
    __shared__ float ldsS1[4][16 * 16];    // 4 KB: per-wave C staging

    const int t = threadIdx.x;

    // Cooperative load of W^T into LDS (heads 8..15 = 0). 24 KB from L2.
    for (int i = t; i < DD * 16; i += 128) {
        int d = i >> 4, col = i & 15;
        ldsW[i] = (col < HH) ? W[col * DD + d] : 0.0f;
    }
    __syncthreads();

    const int wave = t >> 5;
    const int lane = t & 31;
    const int half = lane >> 4;            // 0: lanes 0-15, 1: lanes 16-31
    const int lcol = lane & 15;
    const int tileBase = (blockIdx.x * 4 + wave) * 16;

    // A: lane (half,lcol) supplies row (tileBase+lcol), K = k0 + 2*half {+0,+1}
    const float* aptr = emb + (size_t)(tileBase + lcol) * DD + (half << 1);
    // B: same half-wave K split, column = lcol, from LDS W^T
    const float* bptr = ldsW + ((half << 1) << 4) + lcol;

    v8f c = {0.f, 0.f, 0.f, 0.f, 0.f, 0.f, 0.f, 0.f};

    #pragma unroll 4
    for (int k0 = 0; k0 < DD; k0 += 4) {
        v2f a = *reinterpret_cast<const v2f*>(aptr + k0);   // global_load_b64
        v2f b;
        b.x = bptr[k0 * 16];                                // ds_load_b32
        b.y = bptr[(k0 + 1) * 16];                          // ds_load_b32
        c = __builtin_amdgcn_wmma_f32_16x16x4_f32(
                /*neg_a=*/false, a, /*neg_b=*/false, b,
                /*c_mod=*/(short)0, c, /*reuse_a=*/false, /*reuse_b=*/false);
    }

    // Stage C (S1 tile) to LDS. VGPR r: lanes 0-15 -> M=r, lanes 16-31 -> M=r+8.
    float* sTile = ldsS1[wave];
    {
        const int rbase = half << 3;
        #pragma unroll
        for (int r = 0; r < 8; ++r)
            sTile[(rbase + r) * 16 + lcol] = c[r];
    }
    // Same-wave LDS ops are in-order: reads below see the writes above.

    // Epilogue: 256 outputs per tile, 32 lanes x 8 iterations.
    const size_t maskBase = (size_t)BB * LL;
    #pragma unroll
    for (int it = 0; it < 8; ++it) {
        int idx = it * 32 + lane;
        int m   = idx >> 4;
        int cl  = idx & 15;
        int b   = tileBase + m;
        int p   = pidx[b];
        int h   = head_map[p * LL + cl];
        float logit = sTile[m * 16 + h] + s2[p * LL + cl];
        float pred  = 1.0f / (1.0f + __expf(-logit));
        float msk   = (cl < crit_len[p]) ? 1.0f : 0.0f;
        size_t o = (size_t)b * LL + cl;
        out[o]            = pred * msk;   // contiguous 128B per store
        out[maskBase + o] = msk;
    }
}

// ---------------------------------------------------------------------------
extern "C" void kernel_launch(void* const* d_in, const int* in_sizes, int n_in,
                              void* d_out, int out_size, void* d_ws, size_t ws_size,
                              hipStream_t stream) {
    const float* emb   = (const float*)d_in[0];   // [B, D]
    const float* crit  = (const float*)d_in[1];   // [P, L, D]
    const float* W     = (const float*)d_in[2];   // [H, D]
    const float* bias  = (const float*)d_in[3];   // [H]
    const int*   pidx  = (const int*)d_in[4];     // [B]
    const int*   hmap  = (const int*)d_in[5];     // [P, L]
    const int*   clen  = (const int*)d_in[6];     // [P]
    float*       out   = (float*)d_out;           // [B,L] preds ++ [B,L] mask
    float*       s2    = (float*)d_ws;            // P*L floats (2 KB)

    // Kernel 1: 512 waves -> 16384 threads -> 64 blocks of 256
    mdmc_s2_kernel<<<64, 256, 0, stream>>>(crit, W, bias, hmap, s2);

    // Kernel 2: 1024 wave-tiles / 4 waves per block -> 256 blocks of 128
    mdmc_main_kernel<<<BB / (4 * 16), 128, 0, stream>>>(
        emb, W, pidx, hmap, clen, s2, out);
}